// SampledSkeletonRegressionLoss_83880711290951
// MI455X (gfx1250) — compile-verified
//
#include <hip/hip_runtime.h>
#include <math.h>

#define B_ 4
#define S_ 2
#define P_ 17
#define H_ 128
#define W_ 128
#define M_ 8
#define K_ 10
#define NCH_ 34            // (1+TAG)*P
#define HW_ 16384          // H*W
#define PHW_ 278528        // P*H*W
#define E_INV_ 0.36787944117144233f
#define EPS_ 1e-6f
#define SF_ 3.0f
#define INV2SIG2_ 12.5f    // 1/(2*0.2^2)
#define MAXDIST_ 256.0f    // H+W
#define NEG_INF_ (-1e30f)

typedef __attribute__((ext_vector_type(2))) float v2f;
typedef __attribute__((ext_vector_type(8))) float v8f;

// ---------- wave reductions ----------
__device__ inline float waveSum(float v) {
  #pragma unroll
  for (int m = 16; m > 0; m >>= 1) v += __shfl_xor(v, m, 32);
  return v;
}
__device__ inline float waveMax(float v) {
  #pragma unroll
  for (int m = 16; m > 0; m >>= 1) v = fmaxf(v, __shfl_xor(v, m, 32));
  return v;
}

// 32-lane sum via V_WMMA_F32_16X16X4_F32 (exact f32 FMA path).
// A[m][0]=x_m (lanes 0-15), A[m][2]=x_{m+16} (lanes 16-31), A[*][1]=A[*][3]=0.
// B = ones -> D[m][n] = x_m + x_{m+16}. C/D frag: vgpr j holds rows j (lanes
// 0-15) and j+8 (lanes 16-31), so sum of 8 vgprs + xor-16 add = full sum.
__device__ inline float wmmaSum32(float x) {
  v2f a; a.x = x; a.y = 0.0f;
  v2f b; b.x = 1.0f; b.y = 1.0f;
  v8f c = {};
  c = __builtin_amdgcn_wmma_f32_16x16x4_f32(false, a, false, b, (short)0, c,
                                            false, false);
  float partial = c[0] + c[1] + c[2] + c[3] + c[4] + c[5] + c[6] + c[7];
  partial += __shfl_xor(partial, 16, 32);
  return partial;
}

// ---------- kernel 0: zero accumulators ----------
__global__ void zero_kernel(float* F) {
  if (threadIdx.x < 32) F[threadIdx.x] = 0.0f;
}

// ---------- kernel 1: det (weighted BCE) global mean numerator ----------
__global__ void det_kernel(const float* __restrict__ preds,
                           const float* __restrict__ masks,
                           const float* __restrict__ gh,
                           float* __restrict__ det_sum) {
  const long long n = (long long)B_ * S_ * P_ * HW_;
  long long i = (long long)blockIdx.x * blockDim.x + threadIdx.x;
  float acc = 0.0f;
  if (i < n) {
    int hw = (int)(i % HW_);
    long long t = i / HW_;
    int p = (int)(t % P_); t /= P_;
    int s = (int)(t % S_);
    int b = (int)(t / S_);
    float hmv  = preds[((long long)(b * S_ + s) * NCH_ + p) * HW_ + hw];
    float prob = 1.0f / (1.0f + expf(-hmv));
    float g    = gh[((long long)(b * P_ + p)) * HW_ + hw];
    float tt   = (g > E_INV_) ? 1.0f : 0.0f;
    float wgt  = 10.0f * tt + 0.5f * ((g < E_INV_) ? 1.0f : 0.0f);
    float x    = prob * masks[b * HW_ + hw];
    float lx   = fmaxf(logf(x), -100.0f);
    float l1x  = fmaxf(logf(1.0f - x), -100.0f);
    acc = -wgt * (tt * lx + (1.0f - tt) * l1x);
  }
  acc = waveSum(acc);
  __shared__ float sacc[8];
  int lane = threadIdx.x & 31, wv = threadIdx.x >> 5;
  if (lane == 0) sacc[wv] = acc;
  __syncthreads();
  if (wv == 0) {
    float v = (lane < (int)(blockDim.x >> 5)) ? sacc[lane] : 0.0f;
    v = waveSum(v);
    if (lane == 0) atomicAdd(det_sum, v);
  }
}

// ---------- kernel 2: top-K + tv + person assignment, per (b,s) ----------
__global__ void topk_kernel(const float* __restrict__ preds,
                            const float* __restrict__ skel,
                            float* __restrict__ tv_out,
                            int* __restrict__ person_out) {
  const int bs = blockIdx.x;            // b*S+s
  const int b  = bs / S_;
  const float* hm  = preds + (long long)bs * NCH_ * HW_;          // chans 0..P
  const float* tag = hm + (long long)P_ * HW_;                    // chans P..2P

  float v[K_]; int id[K_];
  #pragma unroll
  for (int j = 0; j < K_; ++j) { v[j] = NEG_INF_; id[j] = 0x7FFFFFFF; }
  for (int i = threadIdx.x; i < PHW_; i += blockDim.x) {
    float val = hm[i];                  // sigmoid is monotone: rank by raw hm
    if (val > v[K_ - 1]) {
      int j = K_ - 1;
      while (j > 0 && val > v[j - 1]) { v[j] = v[j - 1]; id[j] = id[j - 1]; --j; }
      v[j] = val; id[j] = i;
    }
  }
  __shared__ float sv[256 * K_];
  __shared__ int   si[256 * K_];
  #pragma unroll
  for (int j = 0; j < K_; ++j) {
    sv[threadIdx.x * K_ + j] = v[j];
    si[threadIdx.x * K_ + j] = id[j];
  }
  __syncthreads();
  if (threadIdx.x == 0) {
    int topk[K_];
    const int n = 256 * K_;
    for (int k = 0; k < K_; ++k) {
      float best = NEG_INF_; int bi = 0x7FFFFFFF; int pos = 0;
      for (int t = 0; t < n; ++t) {
        float val = sv[t]; int idx = si[t];
        if (val > best || (val == best && idx < bi)) { best = val; bi = idx; pos = t; }
      }
      sv[pos] = NEG_INF_;
      topk[k] = bi;
    }
    for (int k = 0; k < K_; ++k) {
      int idx  = topk[k];
      int part = idx / HW_;
      int rem  = idx - part * HW_;
      float posx = (float)(rem / W_);
      float posy = (float)(rem % W_);
      float tvv  = tag[idx];
      int bestm = 0; float bestd = 1e30f;
      for (int m = 0; m < M_; ++m) {
        const float* sk = skel + ((long long)(b * M_ + m) * P_ + part) * 3;
        float vp = (sk[2] == 1.0f) ? MAXDIST_ : 0.0f;
        float d0 = vp + sk[1] - posx;        // ref uses components [1:3]
        float d1 = vp + sk[2] - posy;
        float dd = sqrtf(d0 * d0 + d1 * d1 + 1e-12f);
        if (dd < bestd) { bestd = dd; bestm = m; }
      }
      int o = bs * K_ + k;
      tv_out[o] = tvv;
      person_out[o] = bestm;
    }
  }
}

// ---------- kernel 3: push/pull tag loss, one thread per (b,s) ----------
__global__ void tag_kernel(const float* __restrict__ preds,
                           const int* __restrict__ kp,
                           float* __restrict__ push_out,
                           float* __restrict__ pull_out) {
  int t = threadIdx.x;
  if (t >= B_ * S_) return;
  int b = t / S_;
  const float* tag = preds + ((long long)t * NCH_ + P_) * HW_;

  float mean_tag[M_], cnt[M_], valid[M_], safe[M_];
  float pull = 0.0f, num = 0.0f;
  for (int m = 0; m < M_; ++m) {
    float g[P_], vis[P_];
    float c = 0.0f, sum = 0.0f;
    for (int p = 0; p < P_; ++p) {
      int base = ((b * M_ + m) * P_ + p) * 2;
      int idx  = kp[base + 0];
      vis[p]   = (kp[base + 1] > 0) ? 1.0f : 0.0f;
      g[p]     = tag[idx];
      c   += vis[p];
      sum += g[p] * vis[p];
    }
    cnt[m]   = c;
    valid[m] = (c > 0.0f) ? 1.0f : 0.0f;
    safe[m]  = fmaxf(c, 1.0f);
    mean_tag[m] = sum / safe[m];
    float pp = 0.0f;
    for (int p = 0; p < P_; ++p) {
      float d = g[p] - mean_tag[m];
      pp += d * d * vis[p];
    }
    pull += (pp / safe[m]) * valid[m];   // TAG==1
    num  += valid[m];
  }
  pull /= (num + EPS_);
  float pmsum = 0.0f;
  for (int i = 0; i < M_; ++i)
    for (int j = 0; j < M_; ++j) {
      float d = mean_tag[i] - mean_tag[j];
      pmsum += expf(-d * d) * valid[i] * valid[j];
    }
  float push = (pmsum - num) / ((num - 1.0f) * num + EPS_) * 0.5f;
  push_out[t] = push;
  pull_out[t] = pull;
}

// ---------- kernel 4: softmax expected-coords + Huber, one wave per task ----
__global__ void __launch_bounds__(32)
softmax_kernel(const float* __restrict__ preds,
               const float* __restrict__ skel,
               const float* __restrict__ tv_in,
               const int* __restrict__ person_in,
               float* __restrict__ hub_sum) {
  int blk = blockIdx.x;                    // (((b*S+s)*K+k)*P + p)
  int p = blk % P_; int r = blk / P_;
  int k = r % K_;   r /= K_;
  int s = r % S_;   int b = r / S_;
  int bs = b * S_ + s;
  const float* hm = preds + ((long long)bs * NCH_ + p) * HW_;
  const float* tg = preds + ((long long)bs * NCH_ + P_ + p) * HW_;
  const float tvv = tv_in[bs * K_ + k];
  const int lane = threadIdx.x;

  // pass 1: max logit
  float mx = NEG_INF_;
  for (int i = lane; i < HW_; i += 32) {
    float d  = tg[i] - tvv;
    float lg = SF_ * hm[i] - INV2SIG2_ * d * d;
    mx = fmaxf(mx, lg);
  }
  mx = waveMax(mx);

  // pass 2: weighted sums (streams hit L2; prefetch next line)
  float s0 = 0.0f, sx = 0.0f, sy = 0.0f;
  for (int i = lane; i < HW_; i += 32) {
    __builtin_prefetch(&hm[i + 256], 0, 1);
    float d  = tg[i] - tvv;
    float e  = expf(SF_ * hm[i] - INV2SIG2_ * d * d - mx);
    s0 += e;
    sx += e * (float)(i >> 7);
    sy += e * (float)(i & 127);
  }
  s0 = wmmaSum32(s0);
  sx = wmmaSum32(sx);
  sy = wmmaSum32(sy);

  if (lane == 0) {
    float px = sx / s0, py = sy / s0;
    int person = person_in[bs * K_ + k];
    const float* sk = skel + ((long long)(b * M_ + person) * P_ + p) * 3;
    float visf = (sk[2] > 0.0f) ? 1.0f : 0.0f;
    float dx = sk[0] - px, dy = sk[1] - py;
    float err = sqrtf(dx * dx + dy * dy + 1e-12f) * visf;
    float hub = (err < 0.1f) ? err * err : 0.1f * (err - 0.05f);
    atomicAdd(hub_sum, hub);
  }
}

// ---------- kernel 5: combine ----------
__global__ void combine_kernel(const float* __restrict__ F,
                               float* __restrict__ out) {
  float pushm = 0.0f, pullm = 0.0f;
  for (int i = 0; i < B_ * S_; ++i) { pushm += F[2 + i]; pullm += F[10 + i]; }
  pushm /= (float)(B_ * S_);
  pullm /= (float)(B_ * S_);
  float det = F[0] / (float)((long long)B_ * S_ * P_ * HW_);
  float skl = 10.0f * F[1] / (float)(B_ * S_ * K_ * P_);
  out[0] = pushm + pullm + det + skl;
}

extern "C" void kernel_launch(void* const* d_in, const int* in_sizes, int n_in,
                              void* d_out, int out_size, void* d_ws,
                              size_t ws_size, hipStream_t stream) {
  const float* preds = (const float*)d_in[0];
  const float* masks = (const float*)d_in[1];
  const float* gh    = (const float*)d_in[2];
  const float* skel  = (const float*)d_in[3];
  const int*   kp    = (const int*)d_in[4];

  // ws layout (floats): [0]=det_sum [1]=hub_sum [2..9]=push [10..17]=pull
  //                     [18..97]=tv ; ints at +128: [0..79]=person
  float* F = (float*)d_ws;
  int*   I = (int*)d_ws + 128;
  float* det_sum = F + 0;
  float* hub_sum = F + 1;
  float* push    = F + 2;
  float* pull    = F + 10;
  float* tv      = F + 18;
  int*   person  = I;

  zero_kernel<<<1, 32, 0, stream>>>(F);

  const long long n_det = (long long)B_ * S_ * P_ * HW_;
  det_kernel<<<(int)((n_det + 255) / 256), 256, 0, stream>>>(preds, masks, gh,
                                                             det_sum);
  topk_kernel<<<B_ * S_, 256, 0, stream>>>(preds, skel, tv, person);
  tag_kernel<<<1, B_ * S_, 0, stream>>>(preds, kp, push, pull);
  softmax_kernel<<<B_ * S_ * K_ * P_, 32, 0, stream>>>(preds, skel, tv, person,
                                                       hub_sum);
  combine_kernel<<<1, 1, 0, stream>>>(F, (float*)d_out);
}